// SAGE_16209206575331
// MI455X (gfx1250) — compile-verified
//
#include <hip/hip_runtime.h>
#include <math.h>

typedef __attribute__((ext_vector_type(2))) float v2f;
typedef __attribute__((ext_vector_type(8))) float v8f;

#define GN0 409600
#define GN1 102400
#define GN2 25600
#define GN3 4096
#define NB 4   // N-tiles (16 cols each) per wave: 64 output columns

// ---------------------------------------------------------------------------
// Zero-fill
// ---------------------------------------------------------------------------
__global__ void k_zero(float* __restrict__ p, int n) {
  int i = blockIdx.x * blockDim.x + threadIdx.x;
  if (i < n) p[i] = 0.0f;
}

// ---------------------------------------------------------------------------
// Edge scatter: one wave32 per edge. Coalesced feature-row copy with
// global float atomics; lane 0 bumps the in-degree count.
// ---------------------------------------------------------------------------
__global__ void k_scatter(const float* __restrict__ x, const int* __restrict__ src,
                          const int* __restrict__ dst, float* __restrict__ sum,
                          float* __restrict__ cnt, int E, int K) {
  int warp = blockIdx.x * (blockDim.x >> 5) + (threadIdx.x >> 5);
  int lane = threadIdx.x & 31;
  if (warp >= E) return;
  int s = src[warp];
  int d = dst[warp];
  const float* xs = x + (size_t)s * K;
  float* sd = sum + (size_t)d * K;
  for (int f = lane; f < K; f += 32) {
    atomicAdd(&sd[f], xs[f]);
  }
  if (lane == 0) atomicAdd(&cnt[d], 1.0f);
}

// ---------------------------------------------------------------------------
// Fused SAGE layer GEMM on the fp32 WMMA pipe:
//   out[m][n] = relu( (sum[m]/max(cnt[m],1)) . Wl[n] + bl[n] + Xt[m] . Wr[n] )
// One wave32 per (16-row M-tile x NB*16-col N-slab).  K stepped by 4 with
// V_WMMA_F32_16X16X4_F32.  A-fragment (16x4): lane%16 = M,
// K = reg + 2*(lane/16) -> one contiguous float2 load per lane, reused
// across the NB column tiles (4x less streamed-activation re-read).
// B-fragment (4x16): same K mapping -> float2 from row-major W[n][k].
// D: reg i, half h -> M = i + 8h, N = lane%16.
// Mean division folded into the A load as a per-row scale.
// ---------------------------------------------------------------------------
__global__ void k_sage_gemm(const float* __restrict__ S, const float* __restrict__ cnt,
                            const float* __restrict__ Xt,
                            const float* __restrict__ Wl, const float* __restrict__ bl,
                            const float* __restrict__ Wr,
                            float* __restrict__ out,
                            int M, int N, int K, int ldOut, int applyRelu) {
  const int tilesM  = M >> 4;
  const int tilesN  = (N + 15) >> 4;
  const int groupsN = (tilesN + NB - 1) / NB;
  int tileId = blockIdx.x * (blockDim.x >> 5) + (threadIdx.x >> 5);
  if (tileId >= tilesM * groupsN) return;  // wave-uniform: EXEC stays all-1s

  int tm = tileId % tilesM;
  int tg = tileId / tilesM;
  int lane = threadIdx.x & 31;
  int l16  = lane & 15;
  int hk   = lane >> 4;                    // lane half selects K+2

  int row = (tm << 4) + l16;               // A-matrix row for this lane
  float inv = 1.0f / fmaxf(cnt[row], 1.0f);
  const float* Srow = S  + (size_t)row * K + (hk << 1);
  const float* Xrow = Xt + (size_t)row * K + (hk << 1);

  int col[NB], colc[NB];
  const float* Wlr[NB];
  const float* Wrr[NB];
#pragma unroll
  for (int j = 0; j < NB; ++j) {
    col[j]  = (((tg * NB) + j) << 4) + l16;       // B/D column for this lane
    colc[j] = (col[j] < N) ? col[j] : (N - 1);    // clamp for safe weight loads
    Wlr[j]  = Wl + (size_t)colc[j] * K + (hk << 1);
    Wrr[j]  = Wr + (size_t)colc[j] * K + (hk << 1);
  }

  v8f c[NB] = {};
  for (int k0 = 0; k0 < K; k0 += 4) {
    v2f a = *(const v2f*)(Srow + k0);
    a *= inv;                              // fold mean = sum / max(cnt,1)
    v2f ax = *(const v2f*)(Xrow + k0);
#pragma unroll
    for (int j = 0; j < NB; ++j) {
      v2f bw = *(const v2f*)(Wlr[j] + k0);
      v2f br = *(const v2f*)(Wrr[j] + k0);
      c[j] = __builtin_amdgcn_wmma_f32_16x16x4_f32(false, a,  false, bw,
                                                   (short)0, c[j], false, false);
      c[j] = __builtin_amdgcn_wmma_f32_16x16x4_f32(false, ax, false, br,
                                                   (short)0, c[j], false, false);
    }
  }

#pragma unroll
  for (int j = 0; j < NB; ++j) {
    float bias = bl[colc[j]];
#pragma unroll
    for (int i = 0; i < 8; ++i) {
      int r = (tm << 4) + i + (hk << 3);
      float v = c[j][i] + bias;
      if (applyRelu) v = fmaxf(v, 0.0f);
      if (col[j] < N) out[(size_t)r * ldOut + col[j]] = v;
    }
  }
}

// ---------------------------------------------------------------------------
// Row-wise log-softmax, one wave32 per row (C <= 64).
// ---------------------------------------------------------------------------
__global__ void k_logsoftmax(const float* __restrict__ h, float* __restrict__ out,
                             int rows, int C) {
  int warp = blockIdx.x * (blockDim.x >> 5) + (threadIdx.x >> 5);
  int lane = threadIdx.x & 31;
  if (warp >= rows) return;
  const float* hr = h + (size_t)warp * C;
  float v0 = (lane      < C) ? hr[lane]      : -INFINITY;
  float v1 = (lane + 32 < C) ? hr[lane + 32] : -INFINITY;
  float m = fmaxf(v0, v1);
  for (int o = 16; o > 0; o >>= 1) m = fmaxf(m, __shfl_xor(m, o, 32));
  float e0 = (lane      < C) ? expf(v0 - m) : 0.0f;
  float e1 = (lane + 32 < C) ? expf(v1 - m) : 0.0f;
  float s = e0 + e1;
  for (int o = 16; o > 0; o >>= 1) s += __shfl_xor(s, o, 32);
  float ls = logf(s);
  float* orow = out + (size_t)warp * C;
  if (lane      < C) orow[lane]      = v0 - m - ls;
  if (lane + 32 < C) orow[lane + 32] = v1 - m - ls;
}

// ---------------------------------------------------------------------------
// Host launcher
// ---------------------------------------------------------------------------
extern "C" void kernel_launch(void* const* d_in, const int* in_sizes, int n_in,
                              void* d_out, int out_size, void* d_ws, size_t ws_size,
                              hipStream_t stream) {
  const float* x   = (const float*)d_in[0];
  const float* Wl0 = (const float*)d_in[1];
  const float* bl0 = (const float*)d_in[2];
  const float* Wr0 = (const float*)d_in[3];
  const float* Wl1 = (const float*)d_in[4];
  const float* bl1 = (const float*)d_in[5];
  const float* Wr1 = (const float*)d_in[6];
  const float* Wl2 = (const float*)d_in[7];
  const float* bl2 = (const float*)d_in[8];
  const float* Wr2 = (const float*)d_in[9];
  const int* src0 = (const int*)d_in[10];
  const int* dst0 = (const int*)d_in[11];
  const int* src1 = (const int*)d_in[12];
  const int* dst1 = (const int*)d_in[13];
  const int* src2 = (const int*)d_in[14];
  const int* dst2 = (const int*)d_in[15];
  const int E0 = 1536000, E1 = 256000, E2 = 40960;

  // Workspace layout (floats), sum/cnt reused across layers:
  //   [0, N1*100)            sum   (layer0: N1*100; layer1: N2*256; layer2: N3*256)
  //   [N1*100, N1*100+N1)    cnt   (layer0: N1; layer1: N2; layer2: N3)
  //   [.., +N1*256)          h1
  float* ws  = (float*)d_ws;
  float* sum = ws;
  float* cnt = ws + (size_t)GN1 * 100;
  float* h1  = cnt + GN1;

  // d_out layout: [log_softmax 4096*47][logits 4096*47][penultimate 25600*256]
  float* outF = (float*)d_out;
  float* lsm  = outF;
  float* h3   = outF + (size_t)GN3 * 47;
  float* h2   = h3   + (size_t)GN3 * 47;

  const int T = 256;                 // 8 waves per block
  const int WPB = T / 32;

  // ---------------- Layer 0: x(409600x100) -> h1(102400x256) ----------------
  {
    int nz = GN1 * 100;
    k_zero<<<(nz + T - 1) / T, T, 0, stream>>>(sum, nz);
    k_zero<<<(GN1 + T - 1) / T, T, 0, stream>>>(cnt, GN1);
    k_scatter<<<(E0 + WPB - 1) / WPB, T, 0, stream>>>(x, src0, dst0, sum, cnt, E0, 100);
    int tiles = (GN1 / 16) * ((256 / 16 + NB - 1) / NB);
    k_sage_gemm<<<(tiles + WPB - 1) / WPB, T, 0, stream>>>(
        sum, cnt, x, Wl0, bl0, Wr0, h1, GN1, 256, 100, 256, 1);
  }

  // ---------------- Layer 1: h1(102400x256) -> h2(25600x256) ----------------
  {
    int nz = GN2 * 256;
    k_zero<<<(nz + T - 1) / T, T, 0, stream>>>(sum, nz);
    k_zero<<<(GN2 + T - 1) / T, T, 0, stream>>>(cnt, GN2);
    k_scatter<<<(E1 + WPB - 1) / WPB, T, 0, stream>>>(h1, src1, dst1, sum, cnt, E1, 256);
    int tiles = (GN2 / 16) * ((256 / 16 + NB - 1) / NB);
    k_sage_gemm<<<(tiles + WPB - 1) / WPB, T, 0, stream>>>(
        sum, cnt, h1, Wl1, bl1, Wr1, h2, GN2, 256, 256, 256, 1);
  }

  // ---------------- Layer 2: h2(25600x256) -> h3(4096x47) -------------------
  {
    int nz = GN3 * 256;
    k_zero<<<(nz + T - 1) / T, T, 0, stream>>>(sum, nz);
    k_zero<<<(GN3 + T - 1) / T, T, 0, stream>>>(cnt, GN3);
    k_scatter<<<(E2 + WPB - 1) / WPB, T, 0, stream>>>(h2, src2, dst2, sum, cnt, E2, 256);
    int tiles = (GN3 / 16) * ((((47 + 15) / 16) + NB - 1) / NB);
    k_sage_gemm<<<(tiles + WPB - 1) / WPB, T, 0, stream>>>(
        sum, cnt, h2, Wl2, bl2, Wr2, h3, GN3, 47, 256, 47, 0);
  }

  // ---------------- log_softmax over logits ---------------------------------
  k_logsoftmax<<<(GN3 + WPB - 1) / WPB, T, 0, stream>>>(h3, lsm, GN3, 47);
}